// MambaBlock_11733850653161
// MI455X (gfx1250) — compile-verified
//
#include <hip/hip_runtime.h>
#include <hip/hip_bf16.h>
#include <cstddef>
#include <cstdint>

typedef __attribute__((ext_vector_type(2))) float v2f;
typedef __attribute__((ext_vector_type(8))) float v8f;
typedef __attribute__((ext_vector_type(4))) unsigned int v4u;
typedef __attribute__((ext_vector_type(8))) unsigned int v8u;

#define LDT 68  // LDS row stride in floats: multiple of 4 (16B-aligned B128 dests),
                // 68 % 64 = 4 banks rotation per row -> conflict-free column reads

enum { EP_SPLIT_SILU = 0, EP_PLAIN = 1, EP_SOFTPLUS_BIAS = 2 };

__device__ __forceinline__ float silu_f(float v) {
  return v / (1.0f + __expf(-v));
}
__device__ __forceinline__ float softplus_f(float v) {
  return (v > 20.0f) ? v : log1pf(__expf(v));
}

// Async global -> LDS copy of 16 bytes (per lane). ASYNCcnt tracked.
__device__ __forceinline__ void async_copy_b128(const void* gptr, void* lptr) {
  const uint32_t lds32 = (uint32_t)(uintptr_t)lptr;     // low 32 bits = LDS byte addr
  const uint64_t ga    = (uint64_t)(uintptr_t)gptr;
  asm volatile("global_load_async_to_lds_b128 %0, %1, off"
               :: "v"(lds32), "v"(ga) : "memory");
}
__device__ __forceinline__ void async_wait_all() {
  asm volatile("s_wait_asynccnt 0x0" ::: "memory");
}

// Generic fp32 WMMA GEMM:  out[m,n] = epilogue( sum_k A[m,k] * W[n,k] )
// A: M x K row-major (stride lda), W: N x K row-major (stride ldw).
// Block: 128 threads = 4 waves, 64x64 C tile; each wave -> 16x64 strip.
// Tiles staged with GLOBAL_LOAD_ASYNC_TO_LDS_B128.
template <int EP>
__global__ __launch_bounds__(128) void gemm_wmma(
    const float* __restrict__ A, int lda,
    const float* __restrict__ W, int ldw,
    float* __restrict__ out0, int ld0,
    float* __restrict__ out1, int ld1,
    const float* __restrict__ bias,
    int M, int N, int K, int split)
{
  __shared__ float As[64 * LDT];
  __shared__ float Ws[64 * LDT];

  const int tid  = threadIdx.x;
  const int lane = tid & 31;
  const int lr   = lane & 15;   // lane row within half-wave
  const int hi   = lane >> 4;   // half-wave select (K pair / M+8 select)
  const int wave = tid >> 5;    // 0..3
  const int m0   = wave * 16;   // wave's M offset inside block tile

  const int nBase = blockIdx.x * 64;
  const int mBase = blockIdx.y * 64;

  const int r0 = tid >> 4;  // loader row group 0..7
  const int c4 = tid & 15;  // loader float4 column 0..15

  v8f acc[4];
#pragma unroll
  for (int j = 0; j < 4; ++j)
#pragma unroll
    for (int e = 0; e < 8; ++e) acc[j][e] = 0.0f;

  for (int kt = 0; kt < K; kt += 64) {
    // ---- async-stage 64x64 A tile and 64x64 W tile straight into LDS ----
#pragma unroll
    for (int it = 0; it < 8; ++it) {
      const int r = r0 + it * 8;
      async_copy_b128(&A[(size_t)(mBase + r) * lda + kt + c4 * 4],
                      &As[r * LDT + c4 * 4]);
      int gn = nBase + r;
      if (gn >= N) gn = N - 1;  // clamp: junk feeds only unstored B columns >= N
      async_copy_b128(&W[(size_t)gn * ldw + kt + c4 * 4],
                      &Ws[r * LDT + c4 * 4]);
    }
    // hint next K tile into cache (lowers to global_prefetch_b8)
    if (kt + 64 < K) {
      __builtin_prefetch(&A[(size_t)(mBase + r0) * lda + kt + 64 + c4 * 4], 0, 1);
      if (nBase + r0 < N)
        __builtin_prefetch(&W[(size_t)(nBase + r0) * ldw + kt + 64 + c4 * 4], 0, 1);
    }
    async_wait_all();
    __syncthreads();

    // ---- 16 K-steps of 4 over staged tiles; 4 WMMAs per step (N subtiles) ----
    // fp32 A-frag layout (ISA 7.12.2): lane<16 -> K={k0,k0+1}, lane>=16 -> K={k0+2,k0+3}
#pragma unroll
    for (int k0 = 0; k0 < 64; k0 += 4) {
      v2f a;
      a.x = As[(m0 + lr) * LDT + k0 + 2 * hi];
      a.y = As[(m0 + lr) * LDT + k0 + 2 * hi + 1];
#pragma unroll
      for (int j = 0; j < 4; ++j) {
        v2f bfr;
        bfr.x = Ws[(j * 16 + lr) * LDT + k0 + 2 * hi];
        bfr.y = Ws[(j * 16 + lr) * LDT + k0 + 2 * hi + 1];
        acc[j] = __builtin_amdgcn_wmma_f32_16x16x4_f32(
            false, a, false, bfr, (short)0, acc[j], false, false);
      }
    }
    __syncthreads();
  }

  // ---- epilogue: C/D layout: VGPR i -> M = i + 8*hi, N = lr (within 16x16) ----
#pragma unroll
  for (int j = 0; j < 4; ++j) {
    const int gcol = nBase + j * 16 + lr;
#pragma unroll
    for (int i = 0; i < 8; ++i) {
      const int grow = mBase + m0 + i + 8 * hi;
      const float v = acc[j][i];
      if constexpr (EP == EP_SPLIT_SILU) {
        if (gcol < split)
          out0[(size_t)grow * ld0 + gcol] = v;
        else
          out1[(size_t)grow * ld1 + (gcol - split)] = silu_f(v);
      } else if constexpr (EP == EP_SOFTPLUS_BIAS) {
        if (gcol < N)
          out0[(size_t)grow * ld0 + gcol] = softplus_f(v + bias[gcol]);
      } else {
        if (gcol < N) out0[(size_t)grow * ld0 + gcol] = v;
      }
    }
  }
}

// Depthwise causal conv1d (k=4) + bias + SiLU. One thread per (b,t,c) element.
__global__ __launch_bounds__(256) void conv_silu_kernel(
    const float* __restrict__ xs, const float* __restrict__ w,
    const float* __restrict__ bias, float* __restrict__ xc, int l, int d)
{
  const size_t i = (size_t)blockIdx.x * blockDim.x + threadIdx.x;
  const int c = (int)(i % d);
  const int t = (int)((i / d) % l);
  float acc = bias[c];
#pragma unroll
  for (int j = 0; j < 4; ++j) {
    const int tp = t - 3 + j;
    if (tp >= 0) acc += xs[i - (size_t)(3 - j) * d] * w[c * 4 + j];
  }
  xc[i] = silu_f(acc);
}

// Selective scan. One thread per (b, channel); 16-state vector in registers.
// Per-timestep B/C vectors (shared by all channels of a batch) are DMA'd into
// LDS by the Tensor Data Mover: a 32x32-float 2D tile of x_dbl (row stride 96,
// cols 64..95), issued by wave 0, completion via s_wait_tensorcnt.
__global__ __launch_bounds__(256) void scan_kernel(
    const float* __restrict__ xc, const float* __restrict__ delta,
    const float* __restrict__ xdbl, const float* __restrict__ A_log,
    const float* __restrict__ res_silu, float* __restrict__ y, int l, int d)
{
  const int c  = blockIdx.x * blockDim.x + threadIdx.x;  // channel
  const int bb = blockIdx.y;                             // batch
  const int tid = threadIdx.x;

  __shared__ float sBC[32 * 32];  // 32 steps x (16 B + 16 C), contiguous rows

  float Arow[16];
#pragma unroll
  for (int nn = 0; nn < 16; ++nn) Arow[nn] = -__expf(A_log[c * 16 + nn]);

  float h[16];
#pragma unroll
  for (int nn = 0; nn < 16; ++nn) h[nn] = 0.0f;

  const uint32_t lds_bc = (uint32_t)(uintptr_t)(void*)sBC;  // LDS byte address

  for (int t0 = 0; t0 < l; t0 += 32) {
    __syncthreads();  // previous chunk fully consumed before DMA overwrites
    if (tid < 32) {   // wave 0 issues the TDM descriptor
      const uint64_t ga =
          (uint64_t)(uintptr_t)&xdbl[(size_t)(bb * l + t0) * 96 + 64];
      // D# group 0: count=1 | lds_addr | global_addr(57b) | type=2
      v4u g0;
      g0.x = 1u;
      g0.y = lds_bc;
      g0.z = (uint32_t)ga;
      g0.w = ((uint32_t)(ga >> 32) & 0x01FFFFFFu) | (2u << 30);
      // D# group 1: data_size=4B; tensor 32x32 (tile==tensor, no OOB);
      // tile 32x32; tensor_dim0_stride=96 floats
      v8u g1;
      g1.s0 = (2u << 16);   // data_size=2 (4 bytes), mask/flags = 0
      g1.s1 = (32u << 16);  // tensor_dim0 = 32
      g1.s2 = (32u << 16);  // tensor_dim1 = 32 (low bits)
      g1.s3 = (32u << 16);  // tile_dim0 = 32
      g1.s4 = 32u;          // tile_dim1 = 32
      g1.s5 = 96u;          // tensor_dim0_stride = 96
      g1.s6 = 0u;
      g1.s7 = 0u;
      // D# groups 2/3: zeros (2D tensor; tile_dim3/4 = 0 -> unused)
      v4u g2;
      g2.x = 0u; g2.y = 0u; g2.z = 0u; g2.w = 0u;
      asm volatile("tensor_load_to_lds %0, %1, %2, %3"
                   :: "s"(g0), "s"(g1), "s"(g2), "s"(g2) : "memory");
      __builtin_amdgcn_s_wait_tensorcnt(0);
    }
    __syncthreads();  // all waves observe DMA'd B/C tile

    for (int tt = 0; tt < 32; ++tt) {
      const int t = t0 + tt;
      const size_t idx = (size_t)(bb * l + t) * d + c;
      const float dv = delta[idx];
      const float u  = xc[idx];
      const float du = dv * u;
      const float* Bt = &sBC[tt * 32];
      const float* Ct = Bt + 16;
      float acc = 0.0f;
#pragma unroll
      for (int nn = 0; nn < 16; ++nn) {
        const float dA = __expf(dv * Arow[nn]);
        h[nn] = dA * h[nn] + du * Bt[nn];
        acc += h[nn] * Ct[nn];
      }
      y[idx] = acc * res_silu[idx];
    }
  }
}

extern "C" void kernel_launch(void* const* d_in, const int* in_sizes, int n_in,
                              void* d_out, int out_size, void* d_ws, size_t ws_size,
                              hipStream_t stream) {
  const float* x      = (const float*)d_in[0];  // (b,l,h)
  const float* W_in   = (const float*)d_in[1];  // (2d,h)
  const float* conv_w = (const float*)d_in[2];  // (d,1,4)
  const float* conv_b = (const float*)d_in[3];  // (d,)
  const float* W_x    = (const float*)d_in[4];  // (r+2n, d) = (96, 2048)
  const float* W_dt   = (const float*)d_in[5];  // (d, r)
  const float* b_dt   = (const float*)d_in[6];  // (d,)
  const float* A_log  = (const float*)d_in[7];  // (d, n)
  const float* W_out  = (const float*)d_in[8];  // (h, d)
  float* out = (float*)d_out;

  const int B = 2, L = 2048, H = 1024, D = 2048, R = 64;
  const int M = B * L;  // 4096

  float* ws    = (float*)d_ws;
  float* XS    = ws;                       // (M, D) pre-conv activations
  float* RES   = XS + (size_t)M * D;       // (M, D) silu(res)
  float* XC    = RES + (size_t)M * D;      // (M, D) conv+silu output
  float* XDBL  = XC + (size_t)M * D;       // (M, 96) dt|B|C
  float* DELTA = XDBL + (size_t)M * 96;    // (M, D)
  float* Y     = DELTA + (size_t)M * D;    // (M, D) gated scan output
  (void)ws_size; (void)in_sizes; (void)n_in; (void)out_size;

  const dim3 blk(128);

  // 1) xr = x @ W_in^T ; split -> XS (raw) and RES (silu)
  gemm_wmma<EP_SPLIT_SILU><<<dim3((2 * D) / 64, M / 64), blk, 0, stream>>>(
      x, H, W_in, H, XS, D, RES, D, nullptr, M, 2 * D, H, /*split=*/D);

  // 2) depthwise causal conv1d + bias + SiLU -> XC
  conv_silu_kernel<<<(M * D) / 256, 256, 0, stream>>>(XS, conv_w, conv_b, XC, L, D);

  // 3) x_dbl = XC @ W_x^T  (N=96, tail-clamped)
  gemm_wmma<EP_PLAIN><<<dim3((96 + 63) / 64, M / 64), blk, 0, stream>>>(
      XC, D, W_x, D, XDBL, 96, nullptr, 0, nullptr, M, 96, D, 96);

  // 4) delta = softplus(dt @ W_dt^T + b_dt)   (dt = first 64 cols of XDBL, lda=96)
  gemm_wmma<EP_SOFTPLUS_BIAS><<<dim3(D / 64, M / 64), blk, 0, stream>>>(
      XDBL, 96, W_dt, R, DELTA, D, nullptr, 0, b_dt, M, D, R, D);

  // 5) selective scan (TDM-staged B/C), fused with output gating y * silu(res) -> Y
  scan_kernel<<<dim3(D / 256, B), 256, 0, stream>>>(XC, DELTA, XDBL, A_log, RES, Y, L, D);

  // 6) out = Y @ W_out^T
  gemm_wmma<EP_PLAIN><<<dim3(H / 64, M / 64), blk, 0, stream>>>(
      Y, D, W_out, D, out, H, nullptr, 0, nullptr, M, H, D, H);
}